// MixedSplineLayer_84086869721844
// MI455X (gfx1250) — compile-verified
//
#include <hip/hip_runtime.h>
#include <hip/hip_bf16.h>

// MixedSplineLayer for MI455X (gfx1250), wave32.
//
// Strategy: all spline slope math (Akima / PCHIP / not-a-knot tridiagonal
// solve / softmax) depends only on the 15-knot coefficient vectors, so a
// 1-thread setup kernel precomputes, per interval, a single weighted cubic
//     y(t) = C0 + C1 t + C2 t^2 + C3 t^3,  t = (x - x0) * inv_h
// for (a) the knots-grid splines {akima, linear-interp, cubic, pchip} folded
// together with their softmax weights, and (b) the t-grid c-spline (weight w2).
// Main kernel is a pure streaming pass: 128 MB total traffic -> ~5.5us at
// 23.3 TB/s. Non-temporal b128 loads/stores, LDS-resident coefficient table
// (ds_load_b128 gathers), global_prefetch one grid-stride ahead.

#define NK 15  // NUM_KNOTS

typedef float v4f __attribute__((ext_vector_type(4)));

__device__ __forceinline__ float sgnf(float v) {
    return (v > 0.0f) ? 1.0f : ((v < 0.0f) ? -1.0f : 0.0f);
}

// ---------------- slope builders (match reference numerics) ----------------

__device__ void akima_slopes(const float* xs, const float* ys, float* d) {
    float mc[NK - 1];
    for (int j = 0; j < NK - 1; ++j) mc[j] = (ys[j + 1] - ys[j]) / (xs[j + 1] - xs[j]);
    float m[NK + 3];
    float ml1 = 2.0f * mc[0] - mc[1];
    float ml0 = 2.0f * ml1 - mc[0];
    float mr0 = 2.0f * mc[NK - 2] - mc[NK - 3];
    float mr1 = 2.0f * mr0 - mc[NK - 2];
    m[0] = ml0; m[1] = ml1;
    for (int j = 0; j < NK - 1; ++j) m[2 + j] = mc[j];
    m[NK + 1] = mr0; m[NK + 2] = mr1;                 // m[0..NK+2]
    float dm[NK + 2];
    for (int j = 0; j < NK + 2; ++j) dm[j] = fabsf(m[j + 1] - m[j]);
    float mx = 0.0f;
    for (int i = 0; i < NK; ++i) {
        float f12 = dm[i + 2] + dm[i];
        if (f12 > mx) mx = f12;
    }
    for (int i = 0; i < NK; ++i) {
        float f1 = dm[i + 2], f2 = dm[i], f12 = f1 + f2;
        if (f12 > 1e-9f * mx)
            d[i] = (f1 * m[i + 1] + f2 * m[i + 2]) / f12;
        else
            d[i] = 0.5f * (m[i + 3] + m[i]);
    }
}

// not-a-knot cubic spline knot slopes; the reference's 15x15 system is
// tridiagonal, so a Thomas solve reproduces it.
__device__ void notaknot_slopes(const float* xs, const float* ys, float* d) {
    float dx[NK - 1], m[NK - 1];
    for (int j = 0; j < NK - 1; ++j) {
        dx[j] = xs[j + 1] - xs[j];
        m[j]  = (ys[j + 1] - ys[j]) / dx[j];
    }
    float a[NK], b[NK], c[NK], r[NK];
    float d0 = xs[2] - xs[0];
    float dN = xs[NK - 1] - xs[NK - 3];
    a[0] = 0.0f; b[0] = dx[1]; c[0] = d0;
    r[0] = ((dx[0] + 2.0f * d0) * dx[1] * m[0] + dx[0] * dx[0] * m[1]) / d0;
    for (int i = 1; i < NK - 1; ++i) {
        a[i] = dx[i];
        b[i] = 2.0f * (dx[i - 1] + dx[i]);
        c[i] = dx[i - 1];
        r[i] = 3.0f * (dx[i] * m[i - 1] + dx[i - 1] * m[i]);
    }
    a[NK - 1] = dN; b[NK - 1] = dx[NK - 3]; c[NK - 1] = 0.0f;
    r[NK - 1] = (dx[NK - 2] * dx[NK - 2] * m[NK - 3] +
                 (2.0f * dN + dx[NK - 2]) * dx[NK - 3] * m[NK - 2]) / dN;
    // Thomas forward sweep
    for (int i = 1; i < NK; ++i) {
        float w = a[i] / b[i - 1];
        b[i] -= w * c[i - 1];
        r[i] -= w * r[i - 1];
    }
    d[NK - 1] = r[NK - 1] / b[NK - 1];
    for (int i = NK - 2; i >= 0; --i) d[i] = (r[i] - c[i] * d[i + 1]) / b[i];
}

__device__ float pchip_edge(float h0, float h1, float m0, float m1) {
    float d = ((2.0f * h0 + h1) * m0 - h0 * m1) / (h0 + h1);
    if (sgnf(d) != sgnf(m0)) d = 0.0f;
    else if ((sgnf(m0) != sgnf(m1)) && (fabsf(d) > 3.0f * fabsf(m0))) d = 3.0f * m0;
    return d;
}

__device__ void pchip_slopes(const float* xs, const float* ys, float* d) {
    float h[NK - 1], m[NK - 1];
    for (int j = 0; j < NK - 1; ++j) {
        h[j] = xs[j + 1] - xs[j];
        m[j] = (ys[j + 1] - ys[j]) / h[j];
    }
    for (int i = 0; i < NK - 2; ++i) {
        float mk = m[i], mk1 = m[i + 1];
        bool cond = (sgnf(mk1) != sgnf(mk)) || (mk1 == 0.0f) || (mk == 0.0f);
        if (cond) {
            d[i + 1] = 0.0f;
        } else {
            float w1 = 2.0f * h[i + 1] + h[i];
            float w2 = h[i + 1] + 2.0f * h[i];
            float ms0 = (mk == 0.0f) ? 1.0f : mk;
            float ms1 = (mk1 == 0.0f) ? 1.0f : mk1;
            d[i + 1] = (w1 + w2) / (w1 / ms0 + w2 / ms1);
        }
    }
    d[0]      = pchip_edge(h[0], h[1], m[0], m[1]);
    d[NK - 1] = pchip_edge(h[NK - 2], h[NK - 3], m[NK - 2], m[NK - 3]);
}

__device__ __forceinline__ void acc_hermite(float wgt, float y0, float y1,
                                            float hd0, float hd1,
                                            float& C0, float& C1, float& C2, float& C3) {
    C0 += wgt * y0;
    C1 += wgt * hd0;
    C2 += wgt * (-3.0f * y0 + 3.0f * y1 - 2.0f * hd0 - hd1);
    C3 += wgt * (2.0f * y0 - 2.0f * y1 + hd0 + hd1);
}

// ---------------- setup kernel: builds the 256-float table in d_ws ---------
// Layout (floats): [0..14] knots, [16..30] t_grid, [32..143] 14 records for
// knots grid, [144..255] 14 records for t grid.
// Record (8 floats, 32 B): {C0, C1, C2, C3, x0, inv_h, 0, 0}.

__global__ void spline_setup_kernel(const float* __restrict__ knots,
                                    const float* __restrict__ cak,
                                    const float* __restrict__ cbs,
                                    const float* __restrict__ ccs,
                                    const float* __restrict__ ccub,
                                    const float* __restrict__ cpc,
                                    const float* __restrict__ wts,
                                    float* __restrict__ ws) {
    if (threadIdx.x != 0 || blockIdx.x != 0) return;

    float xs[NK], ya[NK], yb[NK], yc[NK], ycb[NK], yp[NK];
    for (int i = 0; i < NK; ++i) {
        xs[i]  = knots[i];
        ya[i]  = cak[i];
        yb[i]  = cbs[i];
        yc[i]  = ccs[i];
        ycb[i] = ccub[i];
        yp[i]  = cpc[i];
    }
    // softmax weights
    float w[5], wm = -1e30f, s = 0.0f;
    for (int i = 0; i < 5; ++i) { w[i] = wts[i]; wm = fmaxf(wm, w[i]); }
    for (int i = 0; i < 5; ++i) { w[i] = expf(w[i] - wm); s += w[i]; }
    for (int i = 0; i < 5; ++i) w[i] /= s;

    float da[NK], dcb[NK], dp[NK], dcs[NK], tg[NK];
    akima_slopes(xs, ya, da);
    notaknot_slopes(xs, ycb, dcb);
    pchip_slopes(xs, yp, dp);
    for (int i = 0; i < NK; ++i) tg[i] = (float)i / (float)(NK - 1);
    tg[NK - 1] = 1.0f;
    notaknot_slopes(tg, yc, dcs);

    for (int i = 0; i < NK; ++i) { ws[i] = xs[i]; ws[16 + i] = tg[i]; }
    ws[15] = 0.0f; ws[31] = 0.0f;

    // knots-grid records: akima(w0) + linear(w1) + cubic(w3) + pchip(w4)
    for (int j = 0; j < NK - 1; ++j) {
        float h = xs[j + 1] - xs[j], invh = 1.0f / h;
        float C0 = 0.0f, C1 = 0.0f, C2 = 0.0f, C3 = 0.0f;
        acc_hermite(w[0], ya[j],  ya[j + 1],  h * da[j],  h * da[j + 1],  C0, C1, C2, C3);
        acc_hermite(w[3], ycb[j], ycb[j + 1], h * dcb[j], h * dcb[j + 1], C0, C1, C2, C3);
        acc_hermite(w[4], yp[j],  yp[j + 1],  h * dp[j],  h * dp[j + 1],  C0, C1, C2, C3);
        C0 += w[1] * yb[j];
        C1 += w[1] * (yb[j + 1] - yb[j]);
        int base = 32 + j * 8;
        ws[base + 0] = C0; ws[base + 1] = C1; ws[base + 2] = C2; ws[base + 3] = C3;
        ws[base + 4] = xs[j]; ws[base + 5] = invh; ws[base + 6] = 0.0f; ws[base + 7] = 0.0f;
    }
    // t-grid records: c-spline (w2)
    for (int j = 0; j < NK - 1; ++j) {
        float h = tg[j + 1] - tg[j], invh = 1.0f / h;
        float C0 = 0.0f, C1 = 0.0f, C2 = 0.0f, C3 = 0.0f;
        acc_hermite(w[2], yc[j], yc[j + 1], h * dcs[j], h * dcs[j + 1], C0, C1, C2, C3);
        int base = 144 + j * 8;
        ws[base + 0] = C0; ws[base + 1] = C1; ws[base + 2] = C2; ws[base + 3] = C3;
        ws[base + 4] = tg[j]; ws[base + 5] = invh; ws[base + 6] = 0.0f; ws[base + 7] = 0.0f;
    }
}

// ---------------- main streaming kernel ------------------------------------

__device__ __forceinline__ float eval1(float x, float k0, float k14, const float* sm) {
    x = fminf(fmaxf(x, k0), k14);
    // uniform-grid interval selection (knots = linspace(-1,1,15): h = 1/7;
    // t_grid = linspace(0,1,15): h = 1/14). At exact knot values both
    // candidate intervals agree by spline continuity.
    float fa = (x - k0) * 7.0f;
    int ia = (int)fa; ia = ia < 0 ? 0 : (ia > 13 ? 13 : ia);
    float fb = x * 14.0f;
    int ib = (int)fb; ib = ib < 0 ? 0 : (ib > 13 ? 13 : ib);

    const v4f* rA = (const v4f*)(sm + 32) + ia * 2;
    v4f c = rA[0];          // ds_load_b128
    v4f mA = rA[1];         // {x0, inv_h, 0, 0}
    float t = (x - mA.x) * mA.y;
    float r = fmaf(fmaf(fmaf(c.w, t, c.z), t, c.y), t, c.x);

    const v4f* rB = (const v4f*)(sm + 144) + ib * 2;
    v4f cb = rB[0];
    v4f mB = rB[1];
    float tb = (x - mB.x) * mB.y;
    r = fmaf(fmaf(fmaf(cb.w, tb, cb.z), tb, cb.y), tb, cb.x) + r;
    return r;
}

__global__ __launch_bounds__(256)
void spline_eval_kernel(const float* __restrict__ x,
                        const float* __restrict__ ws,
                        float* __restrict__ out, int n4) {
    __shared__ float sm[256];
    sm[threadIdx.x] = ws[threadIdx.x];   // 1 KB coefficient table -> LDS
    __syncthreads();

    const float k0 = sm[0], k14 = sm[14];
    const v4f* xin = (const v4f*)x;
    v4f* yout = (v4f*)out;

    int stride = gridDim.x * blockDim.x;
    for (int i = blockIdx.x * blockDim.x + threadIdx.x; i < n4; i += stride) {
        // prefetch one grid-stride ahead (global_prefetch_b8 on gfx1250)
        if (i + stride < n4) __builtin_prefetch((const void*)(xin + i + stride), 0, 0);
        v4f v = __builtin_nontemporal_load(xin + i);   // streaming: NT load
        v4f o;
        o.x = eval1(v.x, k0, k14, sm);
        o.y = eval1(v.y, k0, k14, sm);
        o.z = eval1(v.z, k0, k14, sm);
        o.w = eval1(v.w, k0, k14, sm);
        __builtin_nontemporal_store(o, yout + i);      // streaming: NT store
    }
}

// ---------------- launcher --------------------------------------------------

extern "C" void kernel_launch(void* const* d_in, const int* in_sizes, int n_in,
                              void* d_out, int out_size, void* d_ws, size_t ws_size,
                              hipStream_t stream) {
    const float* x     = (const float*)d_in[0];
    const float* knots = (const float*)d_in[1];
    const float* cak   = (const float*)d_in[2];
    const float* cbs   = (const float*)d_in[3];
    const float* ccs   = (const float*)d_in[4];
    const float* ccub  = (const float*)d_in[5];
    const float* cpc   = (const float*)d_in[6];
    const float* wts   = (const float*)d_in[7];
    float* out = (float*)d_out;
    float* ws  = (float*)d_ws;

    spline_setup_kernel<<<1, 1, 0, stream>>>(knots, cak, cbs, ccs, ccub, cpc, wts, ws);

    int n  = in_sizes[0];      // 2048 * 8192
    int n4 = n / 4;
    int threads = 256;
    int blocks  = (n4 + threads - 1) / threads;
    if (blocks > 4096) blocks = 4096;   // grid-stride; ~4 float4 per thread
    spline_eval_kernel<<<blocks, threads, 0, stream>>>(x, ws, out, n4);
}